// Net_with_embedding_27436251087105
// MI455X (gfx1250) — compile-verified
//
#include <hip/hip_runtime.h>

#define N_NODES 50000
#define EDGES   800000
#define BN_EPS  1e-5f

typedef __attribute__((ext_vector_type(2))) float v2f;
typedef __attribute__((ext_vector_type(8))) float v8f;

// ---------------- degree / symmetric norm ----------------
__global__ void k_deg_init(float* deg, int n) {
  int i = blockIdx.x * blockDim.x + threadIdx.x;
  if (i < n) deg[i] = 1.0f;                 // self-loop weight
}
__global__ void k_deg_accum(const int* __restrict__ dst, const float* __restrict__ w,
                            float* deg, int e) {
  int i = blockIdx.x * blockDim.x + threadIdx.x;
  if (i < e) atomicAdd(&deg[dst[i]], w[i]);
}
__global__ void k_deg_inv(float* deg, int n) {
  int i = blockIdx.x * blockDim.x + threadIdx.x;
  if (i < n) { float d = deg[i]; deg[i] = d > 0.f ? rsqrtf(d) : 0.f; }
}

// ---------------- embedding gather (x = node ids) ----------------
__global__ void k_gather(const int* __restrict__ x, const float* __restrict__ emb,
                         float* __restrict__ h, int n) {
  int idx = blockIdx.x * blockDim.x + threadIdx.x;   // n*128 threads
  if (idx >= n * 128) return;
  int i = idx >> 7, f = idx & 127;
  h[idx] = emb[(size_t)x[i] * 128 + f];
}

// ---------------- dense GEMM via V_WMMA_F32_16X16X4_F32 ----------------
// One wave computes one 16x16 tile of C = A(n x K) * B(K x Fout).
// A 16x4 layout: lanes 0-15 -> K = {0,1}, lanes 16-31 -> K = {2,3}; M = lane&15.
// B 4x16 layout mirrored; C/D: VGPR r -> row r (lanes 0-15) / r+8 (lanes 16-31).
__global__ void k_gemm_wmma(const float* __restrict__ A, const float* __restrict__ B,
                            float* __restrict__ C, int n, int K, int Fout) {
  int gw     = (blockIdx.x * blockDim.x + threadIdx.x) >> 5;  // global wave id
  int lane   = threadIdx.x & 31;
  int tilesN = Fout >> 4;
  int tilesM = n >> 4;
  if (gw >= tilesM * tilesN) return;        // wave-uniform guard: EXEC stays all-1s
  int tm   = gw / tilesN;
  int tn   = gw - tm * tilesN;
  int m    = lane & 15;
  int half = lane >> 4;                     // 0: lanes 0-15, 1: lanes 16-31

  const float* Arow = A + (size_t)(tm * 16 + m) * K;
  const float* Bcol = B + tn * 16 + m;

  v8f acc = {};
  for (int k = 0; k < K; k += 4) {
    v2f a, b;
    a.x = Arow[k + 2 * half + 0];
    a.y = Arow[k + 2 * half + 1];
    b.x = Bcol[(size_t)(k + 2 * half + 0) * Fout];
    b.y = Bcol[(size_t)(k + 2 * half + 1) * Fout];
    acc = __builtin_amdgcn_wmma_f32_16x16x4_f32(false, a, false, b,
                                                (short)0, acc, false, false);
  }

  float* Crow = C + (size_t)(tm * 16 + half * 8) * Fout + tn * 16 + m;
#pragma unroll
  for (int r = 0; r < 8; ++r)
    Crow[(size_t)r * Fout] = acc[r];
}

// ---------------- init out = dinv^2 * hW (self-loop) + bias ----------------
__global__ void k_selfloop_bias(const float* __restrict__ hW, const float* __restrict__ dinv,
                                const float* __restrict__ b, float* __restrict__ out,
                                int n, int logF) {
  int F   = 1 << logF;
  int idx = blockIdx.x * blockDim.x + threadIdx.x;
  if (idx >= (n << logF)) return;
  int i = idx >> logF, f = idx & (F - 1);
  float di = dinv[i];
  out[idx] = di * di * hW[idx] + b[f];
}

// ---------------- edge scatter-add: out[dst] += norm * hW[src] ----------------
__global__ void k_agg_edges(const float* __restrict__ hW, const int* __restrict__ src,
                            const int* __restrict__ dst, const float* __restrict__ w,
                            const float* __restrict__ dinv, float* out, int e, int logF) {
  int F   = 1 << logF;
  int idx = blockIdx.x * blockDim.x + threadIdx.x;   // e<<logF <= 102.4M, fits int
  if (idx >= (e << logF)) return;
  int ei = idx >> logF, f = idx & (F - 1);
  int s = src[ei], d = dst[ei];
  float norm = dinv[s] * w[ei] * dinv[d];
  atomicAdd(&out[((size_t)d << logF) + f], norm * hW[((size_t)s << logF) + f]);
}

// ---------------- batch norm (+optional ReLU), one block per channel ----------------
__global__ void k_bn(const float* __restrict__ in, const float* __restrict__ g,
                     const float* __restrict__ be, float* __restrict__ out,
                     int n, int F, int relu) {
  int c   = blockIdx.x;
  int tid = threadIdx.x;
  __shared__ float s_sum[256];
  __shared__ float s_sq[256];
  float sum = 0.f, sq = 0.f;
  for (int i = tid; i < n; i += 256) {
    float v = in[(size_t)i * F + c];
    sum += v; sq += v * v;
  }
  s_sum[tid] = sum; s_sq[tid] = sq;
  __syncthreads();
  for (int s = 128; s > 0; s >>= 1) {
    if (tid < s) { s_sum[tid] += s_sum[tid + s]; s_sq[tid] += s_sq[tid + s]; }
    __syncthreads();
  }
  __shared__ float sh_scale, sh_shift;
  if (tid == 0) {
    float mu  = s_sum[0] / (float)n;
    float var = s_sq[0] / (float)n - mu * mu;     // biased variance
    float inv = rsqrtf(var + BN_EPS);
    float sc  = inv * g[c];
    sh_scale = sc;
    sh_shift = be[c] - mu * sc;
  }
  __syncthreads();
  float sc = sh_scale, sh = sh_shift;
  for (int i = tid; i < n; i += 256) {
    float v = in[(size_t)i * F + c] * sc + sh;
    if (relu) v = fmaxf(v, 0.f);
    out[(size_t)i * F + c] = v;
  }
}

extern "C" void kernel_launch(void* const* d_in, const int* in_sizes, int n_in,
                              void* d_out, int out_size, void* d_ws, size_t ws_size,
                              hipStream_t stream) {
  const int*   x   = (const int*)d_in[0];
  const int*   ei  = (const int*)d_in[1];
  const float* w   = (const float*)d_in[2];
  const float* emb = (const float*)d_in[3];
  const float *W1 = (const float*)d_in[4],  *b1 = (const float*)d_in[5],
              *g1 = (const float*)d_in[6],  *be1 = (const float*)d_in[7];
  const float *W2 = (const float*)d_in[8],  *b2 = (const float*)d_in[9],
              *g2 = (const float*)d_in[10], *be2 = (const float*)d_in[11];
  const float *W3 = (const float*)d_in[12], *b3 = (const float*)d_in[13],
              *g3 = (const float*)d_in[14], *be3 = (const float*)d_in[15];
  const int* src = ei;
  const int* dst = ei + EDGES;

  float* ws   = (float*)d_ws;
  float* dinv = ws;
  float* hA   = dinv + N_NODES;
  float* hW   = hA + (size_t)N_NODES * 128;
  float* agg  = hW + (size_t)N_NODES * 128;

  const int n = N_NODES, e = EDGES;

  // symmetric normalization coefficients (shared by all 3 layers)
  k_deg_init <<<(n + 255) / 256, 256, 0, stream>>>(dinv, n);
  k_deg_accum<<<(e + 255) / 256, 256, 0, stream>>>(dst, w, dinv, e);
  k_deg_inv  <<<(n + 255) / 256, 256, 0, stream>>>(dinv, n);

  // embedding lookup
  k_gather<<<(n * 128 + 255) / 256, 256, 0, stream>>>(x, emb, hA, n);

  struct Layer { const float *W, *b, *g, *be; int Fin, Fout, logF, relu; };
  Layer L[3] = {
    {W1, b1, g1, be1, 128, 128, 7, 1},
    {W2, b2, g2, be2, 128,  64, 6, 1},
    {W3, b3, g3, be3,  64,  32, 5, 0},
  };

  for (int li = 0; li < 3; ++li) {
    int Fin = L[li].Fin, Fout = L[li].Fout, logF = L[li].logF;

    // h @ W  (WMMA f32 16x16x4)
    int waves = (n >> 4) * (Fout >> 4);
    k_gemm_wmma<<<(waves * 32 + 255) / 256, 256, 0, stream>>>(hA, L[li].W, hW, n, Fin, Fout);

    // self-loop term + bias initializes the accumulator
    int nf = n << logF;
    k_selfloop_bias<<<(nf + 255) / 256, 256, 0, stream>>>(hW, dinv, L[li].b, agg, n, logF);

    // edge scatter-add
    int ef = e << logF;
    k_agg_edges<<<(ef + 255) / 256, 256, 0, stream>>>(hW, src, dst, w, dinv, agg, e, logF);

    // batch norm (+ReLU on layers 1,2); layer 3 writes d_out
    float* outp = (li == 2) ? (float*)d_out : hA;
    k_bn<<<Fout, 256, 0, stream>>>(agg, L[li].g, L[li].be, outp, n, Fout, L[li].relu);
  }
}